// CrossInvariantPointAttention_13864154431997
// MI455X (gfx1250) — compile-verified
//
#include <hip/hip_runtime.h>
#include <hip/hip_bf16.h>
#include <math.h>

// ---------------------------------------------------------------------------
// Problem constants (from reference)
// ---------------------------------------------------------------------------
#define LD   512
#define LS   2048
#define C_S  384
#define C_H  16
#define H    12
#define PQ   4
#define PV   8
#define CAT_K 576           // H*(C_H + PV*4)
#define LSTR 2052           // padded LDS row stride (floats) to dodge bank conflicts

typedef float v2f __attribute__((ext_vector_type(2)));
typedef float v8f __attribute__((ext_vector_type(8)));

static __device__ __forceinline__ v8f wmma4(v2f a, v2f b, v8f c) {
  // D = A(16x4,f32) * B(4x16,f32) + C(16x16,f32)
  return __builtin_amdgcn_wmma_f32_16x16x4_f32(
      /*neg_a=*/false, a, /*neg_b=*/false, b,
      /*c_mod=*/(short)0, c, /*reuse_a=*/false, /*reuse_b=*/false);
}

// ---------------------------------------------------------------------------
// Generic f32 WMMA GEMM: C[MxN] = A[MxK] @ B[KxN] (+bias). One wave per tile.
// M,N multiples of 16; K multiple of 4.
// ---------------------------------------------------------------------------
__global__ __launch_bounds__(32)
void gemm_wmma_f32(const float* __restrict__ A, const float* __restrict__ B,
                   float* __restrict__ C, int M, int N, int K,
                   const float* __restrict__ bias) {
  const int lane = threadIdx.x;
  const int half = lane >> 4;
  const int l15  = lane & 15;
  const int m0 = blockIdx.y * 16;
  const int n0 = blockIdx.x * 16;

  v8f acc = {};
  const float* arow = A + (size_t)(m0 + l15) * K;
  for (int k0 = 0; k0 < K; k0 += 4) {
    const int ka = k0 + 2 * half;
    v2f a, b;
    a.x = arow[ka];
    a.y = arow[ka + 1];
    b.x = B[(size_t)ka * N + n0 + l15];
    b.y = B[(size_t)(ka + 1) * N + n0 + l15];
    acc = wmma4(a, b, acc);
  }
  for (int r = 0; r < 8; ++r) {
    const int m = m0 + r + 8 * half;
    const int n = n0 + l15;
    float v = acc[r];
    if (bias) v += bias[n];
    C[(size_t)m * N + n] = v;
  }
}

// ---------------------------------------------------------------------------
// Rigid-frame rotation: out[(l*NPT+pt)*3+c] = R_l @ v + t_l
// Input linear layout from GEMM: lin[l*(3*NPT) + c*NPT + pt]  (coord-major)
// ---------------------------------------------------------------------------
__global__ void rotate_points(const float* __restrict__ lin,
                              const float* __restrict__ Rm,
                              const float* __restrict__ t,
                              float* __restrict__ out,
                              int L, int NPT) {
  const int idx = blockIdx.x * blockDim.x + threadIdx.x;
  if (idx >= L * NPT) return;
  const int l = idx / NPT, pt = idx % NPT;
  const float vx = lin[(size_t)l * (3 * NPT) + 0 * NPT + pt];
  const float vy = lin[(size_t)l * (3 * NPT) + 1 * NPT + pt];
  const float vz = lin[(size_t)l * (3 * NPT) + 2 * NPT + pt];
  const float* R = Rm + (size_t)l * 9;
  float* o = out + (size_t)idx * 3;
  o[0] = R[0] * vx + R[1] * vy + R[2] * vz + t[l * 3 + 0];
  o[1] = R[3] * vx + R[4] * vy + R[5] * vz + t[l * 3 + 1];
  o[2] = R[6] * vx + R[7] * vy + R[8] * vz + t[l * 3 + 2];
}

// ---------------------------------------------------------------------------
// Per-(l,h) squared norms of the PQ query/key points.
// pts layout: [l][h][npts][3], first PQ points used. out[l*H+h].
// ---------------------------------------------------------------------------
__global__ void point_norms(const float* __restrict__ pts,
                            float* __restrict__ out, int L, int npts) {
  const int idx = blockIdx.x * blockDim.x + threadIdx.x;
  if (idx >= L * H) return;
  const int l = idx / H, h = idx % H;
  const float* p = pts + ((size_t)l * H + h) * npts * 3;
  float s = 0.f;
  for (int d = 0; d < PQ * 3; ++d) s += p[d] * p[d];
  out[idx] = s;
}

// ---------------------------------------------------------------------------
// Fused IPA attention. Grid (H, LD/16); one wave per block.
// LDS: 16 x LSTR logits + rowmax[16] + rowsum[16]  (~128.4 KB of 320 KB)
// ---------------------------------------------------------------------------
extern __shared__ float s_lds[];

__global__ __launch_bounds__(32)
void ipa_attention(const float* __restrict__ q_lin,   // [LD][H*16]
                   const float* __restrict__ kv,      // [LS][H*32] (k|v)
                   const float* __restrict__ qpR,     // [LD][H][PQ][3]
                   const float* __restrict__ kvpR,    // [LS][H][12][3] (kpts|vpts)
                   const float* __restrict__ qn,      // [LD][H]
                   const float* __restrict__ kn,      // [LS][H]
                   const float* __restrict__ dst_mask,
                   const float* __restrict__ src_mask,
                   const float* __restrict__ head_w,
                   float* __restrict__ a_out,
                   float* __restrict__ asd_out,
                   float* __restrict__ apts_out,
                   float* __restrict__ cat,           // [LD][CAT_K]; o -> cols [0,192)
                   float* __restrict__ opt_buf) {     // [LD][H][PV*3] global-frame
  const int h    = blockIdx.x;
  const int i0   = blockIdx.y * 16;
  const int lane = threadIdx.x;
  const int half = lane >> 4;
  const int l15  = lane & 15;

  float* rowmax = s_lds + 16 * LSTR;
  float* rowsum = rowmax + 16;

  // softplus(head_w) * sqrt(1/(2*(PQ*9/2))) = softplus * (1/6)
  float hwv = head_w[h];
  hwv = (hwv > 20.f ? hwv : log1pf(__expf(hwv))) * (1.f / 6.f);
  const float scaleQK = 0.17677669529663687f;   // sqrt(1/(2*C_H))

  // Invariant A-fragments: q (K=16 -> 4 steps) and qp (K=12 -> 3 steps)
  v2f aq[4], ap[3];
  {
    const float* qrow = q_lin + (size_t)(i0 + l15) * (H * C_H) + h * C_H;
    for (int t = 0; t < 4; ++t) {
      const int ka = 4 * t + 2 * half;
      aq[t].x = qrow[ka]; aq[t].y = qrow[ka + 1];
    }
    const float* prow = qpR + (size_t)(i0 + l15) * (H * PQ * 3) + h * (PQ * 3);
    for (int t = 0; t < 3; ++t) {
      const int ka = 4 * t + 2 * half;
      ap[t].x = prow[ka]; ap[t].y = prow[ka + 1];
    }
  }
  // Per-lane constants for the 8 D-rows this lane owns
  float qn_r[8], dm_r[8], vmax[8];
  for (int r = 0; r < 8; ++r) {
    const int i = i0 + r + 8 * half;
    qn_r[r] = qn[(size_t)i * H + h];
    dm_r[r] = dst_mask[i];
    vmax[r] = -3.4e38f;
  }

  // ---- Pass 1: logits for all j, stash in LDS, emit a_sd / a_pts ----------
  for (int jt = 0; jt < LS / 16; ++jt) {
    const int j0 = jt * 16;
    const int jn = j0 + l15;

    v8f qk = {};
    {
      const float* kb = kv + (size_t)jn * (H * 2 * C_H) + h * (2 * C_H);
      for (int t = 0; t < 4; ++t) {
        const int ka = 4 * t + 2 * half;
        v2f b; b.x = kb[ka]; b.y = kb[ka + 1];
        qk = wmma4(aq[t], b, qk);
      }
    }
    v8f cx = {};
    {
      const float* kb = kvpR + (size_t)jn * (H * 12 * 3) + h * 36; // k_pts: d 0..11
      for (int t = 0; t < 3; ++t) {
        const int ka = 4 * t + 2 * half;
        v2f b; b.x = kb[ka]; b.y = kb[ka + 1];
        cx = wmma4(ap[t], b, cx);
      }
    }
    const float kn_j = kn[(size_t)jn * H + h];
    const float sm   = src_mask[jn];

    for (int r = 0; r < 8; ++r) {
      const int i = i0 + r + 8 * half;
      const float qks = qk[r] * scaleQK;
      const float pa  = -0.5f * hwv * (qn_r[r] + kn_j - 2.f * cx[r]);
      const float g   = dm_r[r] * sm;
      const size_t off = ((size_t)h * LD + i) * LS + jn;
      asd_out[off]  = qks * g;
      apts_out[off] = pa * g;
      const float lg = qks + pa + 1e9f * (g - 1.f);
      s_lds[(r + 8 * half) * LSTR + jn] = lg;
      vmax[r] = fmaxf(vmax[r], lg);
    }
  }

  // ---- Row max: reduce across the 16 lanes of each half -------------------
  for (int r = 0; r < 8; ++r)
    for (int off = 1; off < 16; off <<= 1)
      vmax[r] = fmaxf(vmax[r], __shfl_xor(vmax[r], off, 32));
  if (l15 == 0)
    for (int r = 0; r < 8; ++r) rowmax[r + 8 * half] = vmax[r];
  __syncthreads();

  // ---- exp + row sum (lane -> row l15, half -> column block) --------------
  {
    const int row = l15;
    const float mx = rowmax[row];
    float s = 0.f;
    float4* lp = (float4*)(s_lds + (size_t)row * LSTR + half * (LS / 2));
    for (int c = 0; c < (LS / 2) / 4; ++c) {
      float4 v = lp[c];
      v.x = __expf(v.x - mx); v.y = __expf(v.y - mx);
      v.z = __expf(v.z - mx); v.w = __expf(v.w - mx);
      lp[c] = v;
      s += v.x + v.y + v.z + v.w;
    }
    s += __shfl_xor(s, 16, 32);
    if (half == 0) rowsum[row] = s;
  }
  __syncthreads();

  // ---- normalize in LDS and write `a` -------------------------------------
  {
    const int row = l15;
    const int i = i0 + row;
    const float inv = 1.f / rowsum[row];
    float4* lp = (float4*)(s_lds + (size_t)row * LSTR + half * (LS / 2));
    float4* ao = (float4*)(a_out + ((size_t)h * LD + i) * LS + half * (LS / 2));
    for (int c = 0; c < (LS / 2) / 4; ++c) {
      float4 v = lp[c];
      v.x *= inv; v.y *= inv; v.z *= inv; v.w *= inv;
      lp[c] = v;
      ao[c] = v;
    }
  }
  __syncthreads();

  // ---- o = A @ V (N=16) and o_pt = A @ Vpts (N=24 -> two tiles) -----------
  v8f ov = {}, op0 = {}, op1 = {};
  for (int jt = 0; jt < LS / 16; ++jt) {
    const int j0 = jt * 16;
    for (int t = 0; t < 4; ++t) {
      const int ka = j0 + 4 * t + 2 * half;     // global j of B rows
      v2f a;
      a.x = s_lds[(size_t)l15 * LSTR + ka];
      a.y = s_lds[(size_t)l15 * LSTR + ka + 1];

      const float* v0 = kv + (size_t)ka * (H * 2 * C_H) + h * (2 * C_H) + C_H;
      const float* v1 = v0 + (H * 2 * C_H);
      v2f bv; bv.x = v0[l15]; bv.y = v1[l15];

      const float* p0 = kvpR + (size_t)ka * (H * 12 * 3) + h * 36 + 12; // v_pts
      const float* p1 = p0 + (H * 12 * 3);
      v2f b0; b0.x = p0[l15]; b0.y = p1[l15];
      v2f b1;
      b1.x = (l15 < 8) ? p0[16 + l15] : 0.f;
      b1.y = (l15 < 8) ? p1[16 + l15] : 0.f;

      ov  = wmma4(a, bv, ov);
      op0 = wmma4(a, b0, op0);
      op1 = wmma4(a, b1, op1);
    }
  }
  for (int r = 0; r < 8; ++r) {
    const int i = i0 + r + 8 * half;
    cat[(size_t)i * CAT_K + h * C_H + l15] = ov[r];            // o block
    opt_buf[((size_t)i * H + h) * (PV * 3) + l15] = op0[r];
    if (l15 < 8)
      opt_buf[((size_t)i * H + h) * (PV * 3) + 16 + l15] = op1[r];
  }
}

// ---------------------------------------------------------------------------
// Inverse-frame transform of o_pt + norms -> cat columns [192, 576)
// ---------------------------------------------------------------------------
__global__ void opt_transform(const float* __restrict__ opt_buf,
                              const float* __restrict__ Rm,
                              const float* __restrict__ t,
                              float* __restrict__ cat) {
  const int idx = blockIdx.x * blockDim.x + threadIdx.x;
  if (idx >= LD * H * PV) return;
  const int i = idx / (H * PV);
  const int hp = idx % (H * PV);            // h*PV + p
  const int h = hp / PV, p = hp % PV;
  const float* g = opt_buf + ((size_t)i * H + h) * (PV * 3) + p * 3;
  const float dx = g[0] - t[i * 3 + 0];
  const float dy = g[1] - t[i * 3 + 1];
  const float dz = g[2] - t[i * 3 + 2];
  const float* R = Rm + (size_t)i * 9;
  const float lx = R[0] * dx + R[3] * dy + R[6] * dz;  // R^T (einsum bikj)
  const float ly = R[1] * dx + R[4] * dy + R[7] * dz;
  const float lz = R[2] * dx + R[5] * dy + R[8] * dz;
  const float nrm = sqrtf(lx * lx + ly * ly + lz * lz + 1e-8f);
  float* c = cat + (size_t)i * CAT_K;
  c[192 + hp] = lx;
  c[288 + hp] = ly;
  c[384 + hp] = lz;
  c[480 + hp] = nrm;
}

// ---------------------------------------------------------------------------
// Host-side orchestration
// ---------------------------------------------------------------------------
extern "C" void kernel_launch(void* const* d_in, const int* in_sizes, int n_in,
                              void* d_out, int out_size, void* d_ws, size_t ws_size,
                              hipStream_t stream) {
  const float* s_dst    = (const float*)d_in[0];
  const float* s_src    = (const float*)d_in[1];
  const float* R_dst    = (const float*)d_in[2];
  const float* t_dst    = (const float*)d_in[3];
  const float* R_src    = (const float*)d_in[4];
  const float* t_src    = (const float*)d_in[5];
  const float* dst_mask = (const float*)d_in[6];
  const float* src_mask = (const float*)d_in[7];
  const float* W_q      = (const float*)d_in[8];
  const float* W_kv     = (const float*)d_in[9];
  const float* W_qp     = (const float*)d_in[10];
  const float* W_kvp    = (const float*)d_in[11];
  const float* W_out    = (const float*)d_in[12];
  const float* b_out    = (const float*)d_in[13];
  const float* head_w   = (const float*)d_in[14];

  float* out = (float*)d_out;
  float* s_upd_out = out;                                   // 512*384
  float* a_out     = out + (size_t)LD * C_S;                // 12*512*2048
  float* asd_out   = a_out + (size_t)H * LD * LS;
  float* apts_out  = asd_out + (size_t)H * LD * LS;

  // Workspace bump allocator (floats)
  float* w = (float*)d_ws;
  float* q_lin   = w; w += (size_t)LD * (H * C_H);          // 512*192
  float* kv_buf  = w; w += (size_t)LS * (H * 2 * C_H);      // 2048*384
  float* qp_lin  = w; w += (size_t)LD * (H * PQ * 3);       // 512*144
  float* kvp_lin = w; w += (size_t)LS * (H * 12 * 3);       // 2048*432
  float* qpR     = w; w += (size_t)LD * (H * PQ * 3);
  float* kvpR    = w; w += (size_t)LS * (H * 12 * 3);
  float* qn      = w; w += (size_t)LD * H;
  float* kn      = w; w += (size_t)LS * H;
  float* opt_buf = w; w += (size_t)LD * H * (PV * 3);
  float* cat     = w; w += (size_t)LD * CAT_K;

  // 1) Projections (f32 WMMA GEMMs)
  gemm_wmma_f32<<<dim3((H * C_H) / 16, LD / 16), 32, 0, stream>>>(
      s_dst, W_q, q_lin, LD, H * C_H, C_S, nullptr);
  gemm_wmma_f32<<<dim3((H * 2 * C_H) / 16, LS / 16), 32, 0, stream>>>(
      s_src, W_kv, kv_buf, LS, H * 2 * C_H, C_S, nullptr);
  gemm_wmma_f32<<<dim3((H * PQ * 3) / 16, LD / 16), 32, 0, stream>>>(
      s_dst, W_qp, qp_lin, LD, H * PQ * 3, C_S, nullptr);
  gemm_wmma_f32<<<dim3((H * 12 * 3) / 16, LS / 16), 32, 0, stream>>>(
      s_src, W_kvp, kvp_lin, LS, H * 12 * 3, C_S, nullptr);

  // 2) Rigid-frame rotations + point norms
  {
    const int nq = LD * (H * PQ);
    rotate_points<<<(nq + 255) / 256, 256, 0, stream>>>(qp_lin, R_dst, t_dst,
                                                        qpR, LD, H * PQ);
    const int nk = LS * (H * 12);
    rotate_points<<<(nk + 255) / 256, 256, 0, stream>>>(kvp_lin, R_src, t_src,
                                                        kvpR, LS, H * 12);
    point_norms<<<(LD * H + 255) / 256, 256, 0, stream>>>(qpR, qn, LD, PQ);
    point_norms<<<(LS * H + 255) / 256, 256, 0, stream>>>(kvpR, kn, LS, 12);
  }

  // 3) Fused attention (LDS-resident softmax, 128.4 KB dynamic LDS)
  {
    const size_t shbytes = (size_t)(16 * LSTR + 32) * sizeof(float);
    ipa_attention<<<dim3(H, LD / 16), 32, shbytes, stream>>>(
        q_lin, kv_buf, qpR, kvpR, qn, kn, dst_mask, src_mask, head_w,
        a_out, asd_out, apts_out, cat, opt_buf);
  }

  // 4) Inverse transform + norms into cat
  opt_transform<<<(LD * H * PV + 255) / 256, 256, 0, stream>>>(opt_buf, R_dst,
                                                               t_dst, cat);

  // 5) Final projection: s_upd = cat @ W_out + b_out
  gemm_wmma_f32<<<dim3(C_S / 16, LD / 16), 32, 0, stream>>>(
      cat, W_out, s_upd_out, LD, C_S, CAT_K, b_out);
}